// FusedMoE_11716670783495
// MI455X (gfx1250) — compile-verified
//
#include <hip/hip_runtime.h>
#include <hip/hip_bf16.h>

// FusedMoE for MI455X (gfx1250, wave32).
// HBM-bound (~151MB fp32 weights, AI ~4 FLOP/B => ~6.5us floor at 23.3 TB/s).
//  - fp32 WMMA (V_WMMA_F32_16X16X4_F32): matches fp32 reference precision
//  - K-split=2 + N-split grids: ~1536/2048 active waves so every WGP streams
//  - register software pipeline (prefetch distance 2, 4 independent WMMA
//    chains) to keep ~6 b64 loads in flight per wave instead of 3
//  - deterministic: unique-writer stores, fixed-order partial reduction

#define E_    16
#define H_    1024
#define F_    768
#define T_    64
#define TOPK_ 2
#define P_    (T_ * TOPK_)   // 128 (token, slot) pairs
#define KS_   2              // K-split factor
#define KB1_  (H_ / KS_)     // 512  K per wave, gate/up
#define KB2_  (F_ / KS_)     // 384  K per wave, down

typedef __attribute__((ext_vector_type(2))) float v2f;
typedef __attribute__((ext_vector_type(4))) float v4f;
typedef __attribute__((ext_vector_type(8))) float v8f;

#define WMMA_F32(A, B, C) \
  __builtin_amdgcn_wmma_f32_16x16x4_f32(false, (A), false, (B), (short)0, (C), false, false)

// ---------------------------------------------------------------------------
// Kernel 1: deterministic routing. Single thread builds per-expert pair lists.
// ---------------------------------------------------------------------------
__global__ void moe_route_kernel(const int* __restrict__ ids,
                                 int* __restrict__ counts,
                                 int* __restrict__ lists) {
  if (threadIdx.x == 0 && blockIdx.x == 0) {
    int c[E_];
#pragma unroll
    for (int e = 0; e < E_; ++e) c[e] = 0;
    for (int p = 0; p < P_; ++p) {
      int e = ids[p] & (E_ - 1);
      lists[e * P_ + c[e]] = p;
      c[e]++;
    }
#pragma unroll
    for (int e = 0; e < E_; ++e) counts[e] = c[e];
  }
}

// ---------------------------------------------------------------------------
// Kernel 2: gate+up partial dots. grid = (2 Ksplit x 6 Nsplit, 8 Mtile, 16 E).
// Each wave owns ONE 16-wide N-tile over a 512-wide K half.
// Pipelined: prefetch distance 2, four independent WMMA chains.
// ---------------------------------------------------------------------------
__global__ __launch_bounds__(256) void moe_gateup_kernel(
    const float* __restrict__ x,          // [T, H]
    const float* __restrict__ w13,        // [E, 2F, H]
    const int* __restrict__ counts,       // [E]
    const int* __restrict__ lists,        // [E, P]
    float* __restrict__ g_part,           // [KS, P, F]
    float* __restrict__ u_part) {         // [KS, P, F]
  const int e      = blockIdx.z;
  const int mtile  = blockIdx.y;
  const int kh     = blockIdx.x & 1;          // K half
  const int nsplit = blockIdx.x >> 1;         // 0..5 -> 8 N-tiles each
  const int count  = counts[e];
  const int base   = mtile * 16;
  if (base >= count) return;                  // uniform exit, EXEC stays full

  __shared__ int s_pair[16];
  __shared__ int s_tok[16];
  if (threadIdx.x < 16) {
    int m = threadIdx.x;
    int p = (base + m < count) ? lists[e * P_ + base + m] : -1;
    s_pair[m] = p;
    s_tok[m]  = (p >= 0) ? (p >> 1) : 0;
  }
  __syncthreads();

  const int lane = threadIdx.x & 31;
  const int wave = threadIdx.x >> 5;
  const int lm   = lane & 15;
  const int hi   = lane >> 4;                 // 0: lanes 0-15, 1: lanes 16-31
  const int kb   = kh * KB1_;

  // A fragment (16x4 f32): lane holds (M=lm, K=k0+2*hi .. +1)
  const float* xrow = x + (size_t)s_tok[lm] * H_ + hi * 2 + kb;
  const int n0 = (nsplit * 8 + wave) * 16;    // this wave's N-tile (0..767)
  const float* w13e  = w13 + (size_t)e * (2 * F_ * H_);
  // B fragment (4x16 f32): lane supplies B[K][N=lm] = w[n0+lm][k]
  const float* w1row = w13e + (size_t)(n0 + lm) * H_ + hi * 2 + kb;
  const float* w3row = w13e + (size_t)(F_ + n0 + lm) * H_ + hi * 2 + kb;

  const v8f z = {0.f, 0.f, 0.f, 0.f, 0.f, 0.f, 0.f, 0.f};
  v8f g0 = z, g1 = z, u0 = z, u1 = z;         // 4 independent chains

  // Software pipeline: prologue loads K-steps 0,1
  v2f a0 = *(const v2f*)(xrow);      v2f b1a = *(const v2f*)(w1row);
  v2f b3a = *(const v2f*)(w3row);
  v2f a1 = *(const v2f*)(xrow + 4);  v2f b1b = *(const v2f*)(w1row + 4);
  v2f b3b = *(const v2f*)(w3row + 4);
  for (int k0 = 0; k0 < KB1_ - 8; k0 += 8) {
    // prefetch next two K-steps before consuming current ones
    v2f a2  = *(const v2f*)(xrow + k0 + 8);
    v2f b1c = *(const v2f*)(w1row + k0 + 8);
    v2f b3c = *(const v2f*)(w3row + k0 + 8);
    v2f a3  = *(const v2f*)(xrow + k0 + 12);
    v2f b1d = *(const v2f*)(w1row + k0 + 12);
    v2f b3d = *(const v2f*)(w3row + k0 + 12);
    g0 = WMMA_F32(a0, b1a, g0);  u0 = WMMA_F32(a0, b3a, u0);
    g1 = WMMA_F32(a1, b1b, g1);  u1 = WMMA_F32(a1, b3b, u1);
    a0 = a2; b1a = b1c; b3a = b3c;
    a1 = a3; b1b = b1d; b3b = b3d;
  }
  g0 = WMMA_F32(a0, b1a, g0);  u0 = WMMA_F32(a0, b3a, u0);
  g1 = WMMA_F32(a1, b1b, g1);  u1 = WMMA_F32(a1, b3b, u1);

  // D fragment: VGPR r -> (M = r + 8*hi, N = n0 + lm)
  float* gdst = g_part + (size_t)kh * P_ * F_;
  float* udst = u_part + (size_t)kh * P_ * F_;
#pragma unroll
  for (int r = 0; r < 8; ++r) {
    int p = s_pair[r + hi * 8];
    if (p >= 0) {
      gdst[(size_t)p * F_ + (n0 + lm)] = g0[r] + g1[r];
      udst[(size_t)p * F_ + (n0 + lm)] = u0[r] + u1[r];
    }
  }
}

// ---------------------------------------------------------------------------
// Kernel 3: h = silu(sum_k g) * (sum_k u). Fixed-order sum -> deterministic.
// ---------------------------------------------------------------------------
__global__ void moe_hcombine_kernel(const float* __restrict__ g_part,
                                    const float* __restrict__ u_part,
                                    float* __restrict__ h_ws) {
  int idx = blockIdx.x * blockDim.x + threadIdx.x;
  if (idx >= P_ * F_) return;
  float g = g_part[idx] + g_part[(size_t)P_ * F_ + idx];
  float u = u_part[idx] + u_part[(size_t)P_ * F_ + idx];
  float s = g / (1.0f + __expf(-g));
  h_ws[idx] = s * u;
}

// ---------------------------------------------------------------------------
// Kernel 4: down-proj partial dots. grid = (2 Ksplit x 8 Nsplit, 8 Mtile, 16 E).
// Stages this K-half of the h tile into padded LDS (conflict-free A reads),
// one 16-wide out N-tile per wave, pipelined with prefetch distance 2.
// ---------------------------------------------------------------------------
__global__ __launch_bounds__(256) void moe_down_kernel(
    const float* __restrict__ w2,         // [E, H, F]
    const int* __restrict__ counts,       // [E]
    const int* __restrict__ lists,        // [E, P]
    const float* __restrict__ h_ws,       // [P, F]
    float* __restrict__ o_part) {         // [KS, P, H]
  const int e      = blockIdx.z;
  const int mtile  = blockIdx.y;
  const int kh     = blockIdx.x & 1;          // K half
  const int nsplit = blockIdx.x >> 1;         // 0..7 -> 8 N-tiles each
  const int count  = counts[e];
  const int base   = mtile * 16;
  if (base >= count) return;

  // 388 mod 64 == 4 -> lanes hit disjoint bank pairs on ds_load_b64
  __shared__ __align__(16) float h_tile[16][KB2_ + 4];  // 24832 B
  __shared__ int s_pair[16];

  if (threadIdx.x < 16) {
    int m = threadIdx.x;
    s_pair[m] = (base + m < count) ? lists[e * P_ + base + m] : -1;
  }
  __syncthreads();

  const int lane = threadIdx.x & 31;
  const int wave = threadIdx.x >> 5;
  const int lm   = lane & 15;
  const int hi   = lane >> 4;
  const int kb   = kh * KB2_;

  // Cooperative coalesced stage of this K-half of h (512B per instruction).
  for (int m = wave; m < 16; m += 8) {
    int p = s_pair[m];
    const float* src = h_ws + (size_t)(p >= 0 ? p : 0) * F_ + kb;
#pragma unroll
    for (int c = 0; c < KB2_; c += 128) {
      v4f v = *(const v4f*)(src + c + lane * 4);
      if (p < 0) v = (v4f){0.f, 0.f, 0.f, 0.f};
      *(v4f*)&h_tile[m][c + lane * 4] = v;
    }
  }
  __syncthreads();

  const int n0 = (nsplit * 8 + wave) * 16;    // this wave's out N-tile
  const float* w2row = w2 + (size_t)e * (H_ * F_) +
                       (size_t)(n0 + lm) * F_ + kb + hi * 2;
  const float* hrow = &h_tile[lm][hi * 2];    // A fragment from LDS

  const v8f z = {0.f, 0.f, 0.f, 0.f, 0.f, 0.f, 0.f, 0.f};
  v8f accA = z, accB = z;                     // even/odd K chains

  v2f b0 = *(const v2f*)(w2row);
  v2f b1 = *(const v2f*)(w2row + 4);
  for (int k0 = 0; k0 < KB2_ - 8; k0 += 8) {
    v2f b2 = *(const v2f*)(w2row + k0 + 8);   // prefetch next two K-steps
    v2f b3 = *(const v2f*)(w2row + k0 + 12);
    v2f a0 = *(const v2f*)(hrow + k0);
    v2f a1 = *(const v2f*)(hrow + k0 + 4);
    accA = WMMA_F32(a0, b0, accA);
    accB = WMMA_F32(a1, b1, accB);
    b0 = b2; b1 = b3;
  }
  {
    v2f a0 = *(const v2f*)(hrow + KB2_ - 8);
    v2f a1 = *(const v2f*)(hrow + KB2_ - 4);
    accA = WMMA_F32(a0, b0, accA);
    accB = WMMA_F32(a1, b1, accB);
  }

  float* odst = o_part + (size_t)kh * P_ * H_;
#pragma unroll
  for (int r = 0; r < 8; ++r) {
    int p = s_pair[r + hi * 8];
    if (p >= 0) odst[(size_t)p * H_ + (n0 + lm)] = accA[r] + accB[r];
  }
}

// ---------------------------------------------------------------------------
// Kernel 5: out[t,i] = sum_a tw[t,a] * (o_part[0][2t+a,i] + o_part[1][2t+a,i])
// Writes every element of d_out; fixed summation order -> deterministic.
// ---------------------------------------------------------------------------
__global__ void moe_reduce_kernel(const float* __restrict__ o_part,
                                  const float* __restrict__ tw,
                                  float* __restrict__ out) {
  int idx = blockIdx.x * blockDim.x + threadIdx.x;
  if (idx >= T_ * H_) return;
  int t = idx >> 10;          // H_ == 1024
  int i = idx & (H_ - 1);
  const float* o0 = o_part;
  const float* o1 = o_part + (size_t)P_ * H_;
  int p0 = t * TOPK_ + 0, p1 = t * TOPK_ + 1;
  float r = tw[p0] * (o0[(size_t)p0 * H_ + i] + o1[(size_t)p0 * H_ + i]) +
            tw[p1] * (o0[(size_t)p1 * H_ + i] + o1[(size_t)p1 * H_ + i]);
  out[idx] = r;
}

extern "C" void kernel_launch(void* const* d_in, const int* in_sizes, int n_in,
                              void* d_out, int out_size, void* d_ws, size_t ws_size,
                              hipStream_t stream) {
  (void)in_sizes; (void)n_in; (void)out_size; (void)ws_size;
  const float* x   = (const float*)d_in[0];   // [64,1024] f32
  const int*   ids = (const int*)d_in[1];     // [64,2] i32
  const float* tw  = (const float*)d_in[2];   // [64,2] f32
  const float* w13 = (const float*)d_in[3];   // [16,1536,1024] f32
  const float* w2  = (const float*)d_in[4];   // [16,1024,768] f32
  float* out = (float*)d_out;                 // [64,1024] f32

  // Workspace: g_part[2*P*F] | u_part[2*P*F] | h[P*F] | o_part[2*P*H] | route
  float* g_part = (float*)d_ws;
  float* u_part = g_part + (size_t)KS_ * P_ * F_;
  float* h_ws   = u_part + (size_t)KS_ * P_ * F_;
  float* o_part = h_ws + (size_t)P_ * F_;
  int*   counts = (int*)(o_part + (size_t)KS_ * P_ * H_);
  int*   lists  = counts + E_;

  moe_route_kernel<<<1, 32, 0, stream>>>(ids, counts, lists);
  moe_gateup_kernel<<<dim3(KS_ * 6, 8, E_), 256, 0, stream>>>(
      x, w13, counts, lists, g_part, u_part);
  moe_hcombine_kernel<<<(P_ * F_ + 255) / 256, 256, 0, stream>>>(
      g_part, u_part, h_ws);
  moe_down_kernel<<<dim3(KS_ * 8, 8, E_), 256, 0, stream>>>(
      w2, counts, lists, h_ws, o_part);
  moe_reduce_kernel<<<(T_ * H_ + 255) / 256, 256, 0, stream>>>(o_part, tw, out);
}